// DepthScalingLayer3D_34007551050357
// MI455X (gfx1250) — compile-verified
//
#include <hip/hip_runtime.h>
#include <hip/hip_bf16.h>
#include <float.h>
#include <stdint.h>

#define HH 2048
#define WW 2048
#define NPIX (HH * WW)
#define CHUNK 8192            // floats per block tile in the TDM rescale pass (32 KB LDS)
#define RED_BLOCKS 1024
#define RED_THREADS 256

typedef unsigned int u32x4 __attribute__((ext_vector_type(4)));
typedef int          i32x4 __attribute__((ext_vector_type(4)));
typedef int          i32x8 __attribute__((ext_vector_type(8)));

// ---- order-preserving float <-> uint encoding (for deterministic atomic min/max) ----
__device__ __forceinline__ unsigned encodeOrd(float f) {
    unsigned u = __float_as_uint(f);
    return (u & 0x80000000u) ? ~u : (u | 0x80000000u);
}
__device__ __forceinline__ float decodeOrd(unsigned e) {
    unsigned u = (e & 0x80000000u) ? (e & 0x7fffffffu) : ~e;
    return __uint_as_float(u);
}

// acc layout (12 uints): [0..2] sparse min xyz, [3..5] sparse max xyz,
//                        [6..8] dense  min xyz, [9..11] dense  max xyz
__device__ __forceinline__ bool slotIsMin(int k) { return (k < 3) || (k >= 6 && k < 9); }

__global__ void init_acc(unsigned* __restrict__ acc) {
    if (threadIdx.x == 0) {
        const unsigned emin = encodeOrd(FLT_MAX);    // identity for min (reference fills +big)
        const unsigned emax = encodeOrd(-FLT_MAX);   // identity for max (reference fills -big)
        #pragma unroll
        for (int k = 0; k < 3; ++k) {
            acc[0 + k] = emin; acc[3 + k] = emax;
            acc[6 + k] = emin; acc[9 + k] = emax;
        }
    }
}

__global__ __launch_bounds__(RED_THREADS)
void minmax_reduce(const float* __restrict__ dense,
                   const float* __restrict__ sparse,
                   const int*   __restrict__ maskp,
                   const float* __restrict__ intr,
                   unsigned*    __restrict__ acc) {
    const float fx = intr[0], fy = intr[1], cx = intr[2], cy = intr[3];
    const float invFx = 1.0f / fx, invFy = 1.0f / fy;

    float vals[12];
    #pragma unroll
    for (int k = 0; k < 12; ++k) vals[k] = slotIsMin(k) ? FLT_MAX : -FLT_MAX;

    const int nvec = NPIX / 4;
    for (int v = blockIdx.x * blockDim.x + threadIdx.x; v < nvec;
         v += gridDim.x * blockDim.x) {
        const float4 s4 = ((const float4*)sparse)[v];   // global_load_b128
        const float4 d4 = ((const float4*)dense)[v];    // global_load_b128
        const int4   m4 = ((const int4*)maskp)[v];      // global_load_b128
        const int j  = v << 2;
        const float yr = (float)(j >> 11) - cy;         // row - cy
        const float xb = (float)(j & (WW - 1)) - cx;    // col - cx (first of 4)

        const float sv[4] = {s4.x, s4.y, s4.z, s4.w};
        const float dv[4] = {d4.x, d4.y, d4.z, d4.w};
        const int   mv[4] = {m4.x, m4.y, m4.z, m4.w};
        #pragma unroll
        for (int c = 0; c < 4; ++c) {
            if (mv[c]) {
                const float xs = xb + (float)c;
                float d  = sv[c];
                float px = xs * d * invFx;
                float py = yr * d * invFy;
                vals[0] = fminf(vals[0], px); vals[3] = fmaxf(vals[3], px);
                vals[1] = fminf(vals[1], py); vals[4] = fmaxf(vals[4], py);
                vals[2] = fminf(vals[2], d ); vals[5] = fmaxf(vals[5], d );
                d  = dv[c];
                px = xs * d * invFx;
                py = yr * d * invFy;
                vals[6] = fminf(vals[6], px); vals[9]  = fmaxf(vals[9],  px);
                vals[7] = fminf(vals[7], py); vals[10] = fmaxf(vals[10], py);
                vals[8] = fminf(vals[8], d ); vals[11] = fmaxf(vals[11], d );
            }
        }
    }

    // ---- wave32 butterfly reduction ----
    #pragma unroll
    for (int off = 16; off > 0; off >>= 1) {
        #pragma unroll
        for (int k = 0; k < 12; ++k) {
            const float o = __shfl_xor(vals[k], off, 32);
            vals[k] = slotIsMin(k) ? fminf(vals[k], o) : fmaxf(vals[k], o);
        }
    }

    // ---- cross-wave via LDS (256 threads = 8 wave32) ----
    __shared__ float red[RED_THREADS / 32][12];
    const int lane = threadIdx.x & 31;
    const int wv   = threadIdx.x >> 5;
    if (lane == 0) {
        #pragma unroll
        for (int k = 0; k < 12; ++k) red[wv][k] = vals[k];
    }
    __syncthreads();
    if (threadIdx.x == 0) {
        #pragma unroll
        for (int w = 1; w < RED_THREADS / 32; ++w) {
            #pragma unroll
            for (int k = 0; k < 12; ++k)
                vals[k] = slotIsMin(k) ? fminf(vals[k], red[w][k])
                                       : fmaxf(vals[k], red[w][k]);
        }
        #pragma unroll
        for (int k = 0; k < 12; ++k) {
            const unsigned e = encodeOrd(vals[k]);
            if (slotIsMin(k)) atomicMin(&acc[k], e);
            else              atomicMax(&acc[k], e);
        }
    }
}

__global__ void finalize(const unsigned* __restrict__ acc,
                         float* __restrict__ ratio_p,
                         float* __restrict__ out) {
    if (threadIdx.x == 0 && blockIdx.x == 0) {
        float ss = 0.0f, sd = 0.0f;
        #pragma unroll
        for (int k = 0; k < 3; ++k) {
            const float ts = decodeOrd(acc[3 + k]) - decodeOrd(acc[0 + k]);
            const float td = decodeOrd(acc[9 + k]) - decodeOrd(acc[6 + k]);
            ss += ts * ts;
            sd += td * td;
        }
        ratio_p[0] = sqrtf(ss) / sqrtf(sd);   // scale_sparse / scale_dense
        out[NPIX]  = 1.0f;                    // std scalar (second tuple element)
    }
}

// Rescale pass: TDM (tensor_load_to_lds) stages each 32KB tile into LDS,
// then all 8 waves scale from LDS and store B128 to the output.
__global__ __launch_bounds__(256)
void scale_tdm(const float* __restrict__ dense,
               const float* __restrict__ ratio_p,
               float* __restrict__ out) {
    __shared__ float tile[CHUNK];

    const unsigned long long ga =
        (unsigned long long)(uintptr_t)dense +
        (unsigned long long)blockIdx.x * (unsigned long long)(CHUNK * 4);

    if ((threadIdx.x >> 5) == 0) {   // wave-uniform: only wave 0 issues the DMA
        const unsigned lds_addr = (unsigned)(uintptr_t)(void*)tile; // low 32 bits = LDS byte offset

        // D# group 0: count=1 (valid), lds_addr, 57-bit global addr, type=2 ("image")
        u32x4 g0 = { 1u,
                     lds_addr,
                     (unsigned)(ga & 0xffffffffull),
                     (((unsigned)(ga >> 32)) & 0x01ffffffu) | (2u << 30) };

        // D# group 1: data_size=4B; tensor_dim0=8192; tensor_dim1=1;
        //             tile_dim0=8192; tile_dim1=1; tensor_dim0_stride=8192
        i32x8 g1 = { (int)0x00020000u,          // [17:16]=2 -> 4-byte elems; wg_mask=0
                     (int)(8192u << 16),        // bits[63:48]  = tensor_dim0[15:0]
                     (int)0x00010000u,          // bits[95:80]  = tensor_dim1[15:0] = 1
                     (int)(8192u << 16),        // bits[127:112]= tile_dim0 = 8192
                     1,                         // bits[143:128]= tile_dim1 = 1
                     8192,                      // bits[191:160]= tensor_dim0_stride[31:0]
                     0, 0 };

        i32x4 gz = { 0, 0, 0, 0 };
#if __clang_major__ >= 23
        i32x8 gz8 = { 0, 0, 0, 0, 0, 0, 0, 0 };
        __builtin_amdgcn_tensor_load_to_lds(g0, g1, gz, gz, gz8, 0);
#else
        __builtin_amdgcn_tensor_load_to_lds(g0, g1, gz, gz, 0);
#endif
        __builtin_amdgcn_s_wait_tensorcnt(0);   // s_wait_tensorcnt 0
    }
    __syncthreads();                             // LDS tile now visible to all waves

    const float r     = ratio_p[0];
    const int   base4 = blockIdx.x * (CHUNK / 4);
    #pragma unroll
    for (int k = threadIdx.x; k < CHUNK / 4; k += 256) {
        float4 t = ((const float4*)tile)[k];     // ds_load_b128
        t.x *= r; t.y *= r; t.z *= r; t.w *= r;
        ((float4*)out)[base4 + k] = t;           // global_store_b128
    }
}

extern "C" void kernel_launch(void* const* d_in, const int* in_sizes, int n_in,
                              void* d_out, int out_size, void* d_ws, size_t ws_size,
                              hipStream_t stream) {
    (void)in_sizes; (void)n_in; (void)out_size; (void)ws_size;
    const float* dense  = (const float*)d_in[0];  // depth_estimations
    const float* sparse = (const float*)d_in[1];  // sparse_depths
    const int*   maskp  = (const int*)d_in[2];    // sparse_depth_masks
    const float* intr   = (const float*)d_in[3];  // intrinsics [fx,fy,cx,cy]
    float*       out    = (float*)d_out;          // [NPIX rescaled] + [1 std]

    unsigned* acc     = (unsigned*)d_ws;                 // 12 encoded min/max slots
    float*    ratio_p = (float*)((char*)d_ws + 64);      // ratio scalar

    init_acc<<<1, 32, 0, stream>>>(acc);
    minmax_reduce<<<RED_BLOCKS, RED_THREADS, 0, stream>>>(dense, sparse, maskp, intr, acc);
    finalize<<<1, 32, 0, stream>>>(acc, ratio_p, out);
    scale_tdm<<<NPIX / CHUNK, 256, 0, stream>>>(dense, ratio_p, out);
}